// EstimateGrassmann_76802605187157
// MI455X (gfx1250) — compile-verified
//
#include <hip/hip_runtime.h>
#include <math.h>

#define DIM    32
#define BATCH  65536
#define BLOCKS 512
#define WPB    8                      // waves per block (256 threads, wave32)
#define NWAVES (BLOCKS * WPB)         // 4096
#define MPW    (BATCH / NWAVES)       // 16 matrices per wave

typedef __attribute__((ext_vector_type(2))) float v2f;
typedef __attribute__((ext_vector_type(8))) float v8f;

// Uniform broadcast from a compile-time-constant lane: lowers to v_readlane_b32
// (SGPR result) instead of ds_bpermute_b32 -- keeps the LDS pipe free for the
// WMMA tile staging.
static __device__ __forceinline__ float bcast(float v, int srcLane) {
  return __int_as_float(__builtin_amdgcn_readlane(__float_as_int(v), srcLane));
}

// ---------------------------------------------------------------------------
// WMMA f32 16x16x4 operand helpers. All tiles stored ROW-MAJOR in LDS with
// stride S floats. Layouts per CDNA5 ISA 7.12.2:
//   A (16x4):  lane l holds row M=l%16; VGPR v -> K = v + (l>=16 ? 2 : 0)
//   B (4x16):  lane l holds col N=l%16; VGPR v -> K = v + (l>=16 ? 2 : 0)
//   C/D (16x16): lane l col N=l%16; VGPR r -> M = r + (l>=16 ? 8 : 0)
// ---------------------------------------------------------------------------
static __device__ __forceinline__ v2f lds_loadA(const float* p, int lane, int Mo,
                                                int kb, int S) {
  int r = (lane & 15) + Mo;
  int k = kb + ((lane >> 4) << 1);
  v2f a;
  a.x = p[r * S + k];
  a.y = p[r * S + k + 1];
  return a;
}

static __device__ __forceinline__ v2f lds_loadB(const float* p, int lane, int No,
                                                int kb, int S) {
  int n = (lane & 15) + No;
  int k = kb + ((lane >> 4) << 1);
  v2f b;
  b.x = p[k * S + n];
  b.y = p[(k + 1) * S + n];
  return b;
}

static __device__ __forceinline__ v8f lds_loadC(const float* p, int lane, int Mo,
                                                int No, int S) {
  int n  = (lane & 15) + No;
  int r0 = Mo + ((lane >> 4) << 3);
  v8f c;
#pragma unroll
  for (int r = 0; r < 8; ++r) c[r] = p[(r0 + r) * S + n];
  return c;
}

static __device__ __forceinline__ void lds_storeC(float* p, int lane, int Mo,
                                                  int No, int S, v8f c) {
  int n  = (lane & 15) + No;
  int r0 = Mo + ((lane >> 4) << 3);
#pragma unroll
  for (int r = 0; r < 8; ++r) p[(r0 + r) * S + n] = c[r];
}

// acc += A(16xK) * B(Kx16), K in {16,32}, via V_WMMA_F32_16X16X4_F32
static __device__ __forceinline__ v8f wmma_block(const float* A, const float* B,
                                                 int lane, int Mo, int No, int S,
                                                 int K, v8f acc) {
#pragma unroll
  for (int kb = 0; kb < K; kb += 4) {
    v2f a = lds_loadA(A, lane, Mo, kb, S);
    v2f b = lds_loadB(B, lane, No, kb, S);
    acc = __builtin_amdgcn_wmma_f32_16x16x4_f32(false, a, false, b, (short)0, acc,
                                                false, false);
  }
  return acc;
}

// ---------------------------------------------------------------------------
// Kernel 1: compute sigma (one wave; perf irrelevant, runs once).
// LDS-resident 32x32 ops, lane j works element j of each row.
// ---------------------------------------------------------------------------
static __device__ void stabilize32(float* M, int lane) {
  __syncthreads();
  // diag -> relu(diag); lane handles row==lane's diagonal element
  float d = fmaxf(M[lane * 33], 0.0f);
  __syncthreads();
  M[lane * 33] = d;
  __syncthreads();
  // diag -> rowsum of |M_| (includes the relu'd diag)
  float r = 0.0f;
  for (int j = 0; j < 32; ++j) r += fabsf(M[lane * 32 + j]);
  __syncthreads();
  M[lane * 33] = r;
  __syncthreads();
}

// Gauss-Jordan: A -> I (destroyed), V (init I) -> A^{-1}. Row-major, no pivoting.
static __device__ void gj_invert32(float* A, float* V, int lane) {
  for (int k = 0; k < 32; ++k) {
    __syncthreads();
    float rp = 1.0f / A[k * 33];
    __syncthreads();
    A[k * 32 + lane] *= rp;
    V[k * 32 + lane] *= rp;
    __syncthreads();
    for (int i = 0; i < 32; ++i) {
      if (i == k) continue;
      float m = A[i * 32 + k];
      __syncthreads();
      A[i * 32 + lane] -= m * A[k * 32 + lane];
      V[i * 32 + lane] -= m * V[k * 32 + lane];
      __syncthreads();
    }
  }
  __syncthreads();
}

__global__ __launch_bounds__(32) void sigma_kernel(const float* __restrict__ B,
                                                   const float* __restrict__ C,
                                                   float* __restrict__ sigCol) {
  __shared__ float Bs[1024], Cs[1024], Vc[1024], Lam[1024], Vl[1024];
  int lane = threadIdx.x;
  for (int i = 0; i < 32; ++i) {
    Bs[i * 32 + lane] = B[i * 32 + lane];
    Cs[i * 32 + lane] = C[i * 32 + lane];
    Vc[i * 32 + lane] = (i == lane) ? 1.0f : 0.0f;
    Vl[i * 32 + lane] = (i == lane) ? 1.0f : 0.0f;
  }
  __syncthreads();
  stabilize32(Bs, lane);
  stabilize32(Cs, lane);
  gj_invert32(Cs, Vc, lane);  // Vc = inv(C_), row-major

  // lam = B_ @ inv(C_)  (4 blocks x 8 K-chunks of V_WMMA_F32_16X16X4_F32)
  __syncthreads();
#pragma unroll
  for (int Mo = 0; Mo < 32; Mo += 16)
#pragma unroll
    for (int No = 0; No < 32; No += 16) {
      v8f acc = {};
      acc = wmma_block(Bs, Vc, lane, Mo, No, 32, 32, acc);
      lds_storeC(Lam, lane, Mo, No, 32, acc);
    }
  __syncthreads();
  Lam[lane * 33] += 1.0f;  // + I
  __syncthreads();
  gj_invert32(Lam, Vl, lane);  // Vl = sigma, row-major
  __syncthreads();
  // write sigma column-major: sigCol[j*32 + i] = sigma[i][j]
  for (int i = 0; i < 32; ++i) sigCol[lane * 32 + i] = Vl[i * 32 + lane];
}

// ---------------------------------------------------------------------------
// Kernel 2: batched 32x32 determinants, one matrix per wave iteration.
// Lane j owns column j in registers. Blocked LU: 16-wide panel + unit-lower
// tri-solve + WMMA Schur update + 16x16 tail LU.
// Broadcasts use v_readlane (uniform lane index), not LDS permutes.
// ---------------------------------------------------------------------------
__global__ __launch_bounds__(256) void det_kernel(const int* __restrict__ x,
                                                  const float* __restrict__ sigCol,
                                                  float* __restrict__ partials) {
  __shared__ float sigS[1024];
  __shared__ float stage[WPB][3 * 256];  // per-wave: [-L21 | U12 | D/S], 16x16 each
  const int tid = threadIdx.x, lane = tid & 31, w = tid >> 5;
  for (int i = tid; i < 1024; i += 256) sigS[i] = sigCol[i];
  __syncthreads();
  float* Lb = &stage[w][0];
  float* Ub = &stage[w][256];
  float* Db = &stage[w][512];
  const int waveG = blockIdx.x * WPB + w;

  float acc_log = 0.0f;
  for (int m = 0; m < MPW; ++m) {
    const int b = waveG * MPW + m;
    if (m + 1 < MPW) __builtin_prefetch(&x[(b + 1) * 32], 0, 0);
    // N = sigma - diag(1-x): lane holds column `lane`
    float e = 1.0f - (float)x[b * 32 + lane];
    float n[32];
#pragma unroll
    for (int i = 0; i < 32; ++i)
      n[i] = sigS[lane * 32 + i] - ((i == lane) ? e : 0.0f);

    float prod = 1.0f;
    // ---- panel: factor columns 0..15 over all 32 rows ----
#pragma unroll
    for (int k = 0; k < 16; ++k) {
      float piv = bcast(n[k], k);
      prod *= piv;
      float rp = 1.0f / piv;
      float f  = n[k] * rp;
      bool isK = (lane == k);
      bool upd = (lane > k) && (lane < 16);
#pragma unroll
      for (int i = k + 1; i < 32; ++i) {
        float s  = bcast(n[i], k);       // column k, row i (broadcast)
        float nv = n[i] - f * s;         // eliminate in columns k+1..15
        float lv = s * rp;               // normalized L stored back in column k
        n[i] = isK ? lv : (upd ? nv : n[i]);
      }
    }
    // ---- tri-solve: U12 = L11^{-1} B on columns 16..31, rows 0..15 ----
#pragma unroll
    for (int k = 0; k < 16; ++k) {
#pragma unroll
      for (int i = k + 1; i < 16; ++i) {
        float s  = bcast(n[i], k);       // L11[i][k]
        float nv = n[i] - s * n[k];
        n[i] = (lane >= 16) ? nv : n[i];
      }
    }
    // ---- stage tiles to LDS (row-major, stride 16); L21 pre-negated ----
    if (lane < 16) {
#pragma unroll
      for (int r = 0; r < 16; ++r) Lb[r * 16 + lane] = -n[16 + r];
    } else {
      int c = lane - 16;
#pragma unroll
      for (int i = 0; i < 16; ++i) Ub[i * 16 + c] = n[i];
#pragma unroll
      for (int i = 0; i < 16; ++i) Db[i * 16 + c] = n[16 + i];
    }
    asm volatile("s_wait_dscnt 0" ::: "memory");

    // ---- Schur: S = D + (-L21)*U12 via 4x V_WMMA_F32_16X16X4_F32 ----
    v8f S = lds_loadC(Db, lane, 0, 0, 16);
    S = wmma_block(Lb, Ub, lane, 0, 0, 16, 16, S);
    lds_storeC(Db, lane, 0, 0, 16, S);
    asm volatile("s_wait_dscnt 0" ::: "memory");

    // ---- 16x16 tail LU on S (lanes 16..31 mirror lanes 0..15) ----
    const int cl = lane & 15;
    float s16[16];
#pragma unroll
    for (int i = 0; i < 16; ++i) s16[i] = Db[i * 16 + cl];
#pragma unroll
    for (int k = 0; k < 16; ++k) {
      float piv = bcast(s16[k], k);
      prod *= piv;
      float f = s16[k] / piv;
#pragma unroll
      for (int i = k + 1; i < 16; ++i) {
        float s  = bcast(s16[i], k);
        float nv = s16[i] - f * s;
        s16[i] = (cl > k) ? nv : s16[i];
      }
    }
    // p_b > 0 and det(m_b) = +/-det(N_b)  =>  log p_b = log|det N_b|
    acc_log += __logf(fabsf(prod));
  }
  if (lane == 0) partials[waveG] = acc_log * (1.0f / (float)BATCH);
}

// ---------------------------------------------------------------------------
// Kernel 3: deterministic reduction of wave partials (single wave).
// ---------------------------------------------------------------------------
__global__ __launch_bounds__(32) void reduce_kernel(const float* __restrict__ partials,
                                                    float* __restrict__ out, int n) {
  int lane = threadIdx.x;
  float s = 0.0f;
  for (int i = lane; i < n; i += 32) s += partials[i];
#pragma unroll
  for (int off = 16; off > 0; off >>= 1) s += __shfl_xor(s, off, 32);
  if (lane == 0) out[0] = s;
}

extern "C" void kernel_launch(void* const* d_in, const int* in_sizes, int n_in,
                              void* d_out, int out_size, void* d_ws, size_t ws_size,
                              hipStream_t stream) {
  (void)in_sizes; (void)n_in; (void)out_size; (void)ws_size;
  const int*   x = (const int*)d_in[0];
  const float* B = (const float*)d_in[1];
  const float* C = (const float*)d_in[2];
  float* out      = (float*)d_out;
  float* sigCol   = (float*)d_ws;          // 1024 floats: sigma, column-major
  float* partials = sigCol + 1024;         // NWAVES floats

  sigma_kernel<<<1, 32, 0, stream>>>(B, C, sigCol);
  det_kernel<<<BLOCKS, 256, 0, stream>>>(x, sigCol, partials);
  reduce_kernel<<<1, 32, 0, stream>>>(partials, out, NWAVES);
}